// MultiViewRecommender_7885559955680
// MI455X (gfx1250) — compile-verified
//
#include <hip/hip_runtime.h>
#include <hip/hip_bf16.h>

// ---------------------------------------------------------------------------
// MultiViewRecommender forward for MI455X (gfx1250, wave32).
//
// Bandwidth-bound (~1.3 GB traffic, ~0.6 GFLOP). fp32 throughout.
// Row-wise cosine dots are computed 16 rows at a time as the diagonal of a
// 16x16 Gram block via V_WMMA_F32_16X16X4_F32 (K accumulated 4 at a time).
// ---------------------------------------------------------------------------

typedef float v2f __attribute__((ext_vector_type(2)));
typedef float v4f __attribute__((ext_vector_type(4)));
typedef float v8f __attribute__((ext_vector_type(8)));

#define EMB_D 256
#define EPS_F 1e-8f
#define CL_WEIGHT_F 0.05f

// Branchless extract of component m (0..7) from an 8-wide accumulator.
// m is per-lane (divergent); 3-level mux tree lowers to v_cndmask chain,
// avoiding the exec-mask branch tree clang generates for a ternary ladder.
__device__ __forceinline__ float diag_get(v8f c, int m) {
    bool b0 = (m & 1) != 0;
    bool b1 = (m & 2) != 0;
    bool b2 = (m & 4) != 0;
    float a0 = b0 ? c[1] : c[0];
    float a1 = b0 ? c[3] : c[2];
    float a2 = b0 ? c[5] : c[4];
    float a3 = b0 ? c[7] : c[6];
    float c0 = b1 ? a1 : a0;
    float c1 = b1 ? a3 : a2;
    return b2 ? c1 : c0;
}

__global__ void zero_loss_kernel(float* p) {
    if (threadIdx.x == 0 && blockIdx.x == 0) *p = 0.0f;
}

// One wave32 handles one 16-row block. 8 waves per 256-thread workgroup.
// nb16 = N/16 (N divisible by 16 for all three calls: 200000, 100000, 8192).
__global__ __launch_bounds__(256) void infonce_wmma_kernel(
    const float* __restrict__ e1, const float* __restrict__ e2,
    const int* __restrict__ neg_idx, int nb16, float inv_n,
    float* __restrict__ loss_out) {

    const int wid = threadIdx.x >> 5;
    const int rb  = blockIdx.x * 8 + wid;
    if (rb >= nb16) return;                 // wave-uniform exit; EXEC stays all-1s

    const int lane  = threadIdx.x & 31;
    const int row_l = lane & 15;            // matrix row / col for this lane
    const int half  = lane >> 4;            // K sub-offset selector
    const int row   = rb * 16 + row_l;

    // A (16x4 MxK) and B (4x16 KxN) f32 fragments share the per-lane layout:
    // lane L holds float2 at (row = L&15, k = kbase + (L>>4)*2).
    const float* p1 = e1 + (size_t)row * EMB_D + half * 2;
    const float* p2 = e2 + (size_t)row * EMB_D + half * 2;
    const int    j  = neg_idx[row];
    const float* pg = e2 + (size_t)j * EMB_D + half * 2;
    __builtin_prefetch(pg, 0, 3);           // gathered row: random access, prefetch

    v8f cpos = {};  // diag -> e1_m . e2_m
    v8f cneg = {};  // diag -> e1_m . e2[neg]_m
    v8f cn1  = {};  // diag -> ||e1_m||^2
    v8f cn2  = {};  // diag -> ||e2_m||^2
    v8f cng  = {};  // diag -> ||e2[neg]_m||^2

#pragma unroll 4
    for (int k = 0; k < EMB_D; k += 4) {
        v2f a = *(const v2f*)(p1 + k);
        v2f b = *(const v2f*)(p2 + k);
        v2f g = *(const v2f*)(pg + k);
        cpos = __builtin_amdgcn_wmma_f32_16x16x4_f32(false, a, false, b, (short)0, cpos, false, false);
        cneg = __builtin_amdgcn_wmma_f32_16x16x4_f32(false, a, false, g, (short)0, cneg, false, false);
        cn1  = __builtin_amdgcn_wmma_f32_16x16x4_f32(false, a, false, a, (short)0, cn1,  false, false);
        cn2  = __builtin_amdgcn_wmma_f32_16x16x4_f32(false, b, false, b, (short)0, cn2,  false, false);
        cng  = __builtin_amdgcn_wmma_f32_16x16x4_f32(false, g, false, g, (short)0, cng,  false, false);
    }

    // Diagonal ownership in the 16x16 f32 C/D layout:
    //   rows 0..7  -> lane m      , component m
    //   rows 8..15 -> lane m + 16 , component m - 8
    const bool owns = (lane < 8) || (lane >= 24);
    const int  comp = lane & 7;             // correct component for owning lanes

    float val = 0.0f;
    if (owns) {
        float dp = diag_get(cpos, comp);
        float dn = diag_get(cneg, comp);
        float q1 = diag_get(cn1,  comp);
        float q2 = diag_get(cn2,  comp);
        float qg = diag_get(cng,  comp);
        float r1 = fmaxf(sqrtf(q1), EPS_F);
        float r2 = fmaxf(sqrtf(q2), EPS_F);
        float rg = fmaxf(sqrtf(qg), EPS_F);
        float pos = dp / (r1 * r2);
        float neg = dn / (r1 * rg);
        float x = neg - pos;                       // temperature = 1
        val = (x > 20.0f) ? x : log1pf(__expf(x)); // softplus
    }

    // wave32 reduction, then one atomic per wave
    for (int off = 16; off > 0; off >>= 1) val += __shfl_xor(val, off, 32);
    if (lane == 0) atomicAdd(loss_out, CL_WEIGHT_F * inv_n * val);
}

__device__ __forceinline__ v4f sigmoid4(v4f x) {
    v4f r;
    r[0] = 1.0f / (1.0f + __expf(-x[0]));
    r[1] = 1.0f / (1.0f + __expf(-x[1]));
    r[2] = 1.0f / (1.0f + __expf(-x[2]));
    r[3] = 1.0f / (1.0f + __expf(-x[3]));
    return r;
}

// user_emb[b] = sig(w[0]) * graph_user_emb[user[b]] + sig(w[1]) * seq_last_emb[b]
__global__ __launch_bounds__(256) void fusion_user_kernel(
    const float* __restrict__ gu, const float* __restrict__ seq,
    const float* __restrict__ w, const int* __restrict__ user,
    float* __restrict__ out, int nvec) {
    int idx = blockIdx.x * blockDim.x + threadIdx.x;
    if (idx >= nvec) return;
    int b  = idx >> 6;           // D/4 = 64 float4 per row
    int dv = idx & 63;
    v4f s0 = sigmoid4(*(const v4f*)(w + dv * 4));
    v4f s1 = sigmoid4(*(const v4f*)(w + EMB_D + dv * 4));
    int u  = user[b];
    v4f a  = *(const v4f*)(gu + (size_t)u * EMB_D + dv * 4);
    v4f s  = *(const v4f*)(seq + (size_t)b * EMB_D + dv * 4);
    *(v4f*)(out + (size_t)b * EMB_D + dv * 4) = s0 * a + s1 * s;
}

// item_emb[i] = sig(w[0]) * graph_item_emb[i] + sig(w[1]) * seq_item_emb[i]
__global__ __launch_bounds__(256) void fusion_item_kernel(
    const float* __restrict__ gi, const float* __restrict__ si,
    const float* __restrict__ w, float* __restrict__ out, int nvec) {
    int idx = blockIdx.x * blockDim.x + threadIdx.x;
    if (idx >= nvec) return;
    int dv = idx & 63;
    v4f s0 = sigmoid4(*(const v4f*)(w + dv * 4));
    v4f s1 = sigmoid4(*(const v4f*)(w + EMB_D + dv * 4));
    v4f a  = *(const v4f*)(gi + (size_t)idx * 4);
    v4f s  = *(const v4f*)(si + (size_t)idx * 4);
    *(v4f*)(out + (size_t)idx * 4) = s0 * a + s1 * s;
}

extern "C" void kernel_launch(void* const* d_in, const int* in_sizes, int n_in,
                              void* d_out, int out_size, void* d_ws, size_t ws_size,
                              hipStream_t stream) {
    const float* graph_user_emb    = (const float*)d_in[0];
    const float* graph_user_emb_cl = (const float*)d_in[1];
    const float* graph_item_emb    = (const float*)d_in[2];
    const float* graph_item_emb_cl = (const float*)d_in[3];
    const float* seq_last_emb      = (const float*)d_in[4];
    const float* seq_last_emb_cl   = (const float*)d_in[5];
    const float* seq_item_emb      = (const float*)d_in[6];
    const float* user_attn_w       = (const float*)d_in[7];
    const float* item_attn_w       = (const float*)d_in[8];
    const int*   user              = (const int*)d_in[9];
    const int*   neg_user          = (const int*)d_in[10];
    const int*   neg_item          = (const int*)d_in[11];
    const int*   neg_seq           = (const int*)d_in[12];

    const int U = in_sizes[0] / EMB_D;   // 200000
    const int I = in_sizes[2] / EMB_D;   // 100000
    const int B = in_sizes[4] / EMB_D;   // 8192

    float* out      = (float*)d_out;
    float* user_out = out;
    float* item_out = out + (size_t)B * EMB_D;
    float* loss_out = out + (size_t)B * EMB_D + (size_t)I * EMB_D;

    zero_loss_kernel<<<1, 32, 0, stream>>>(loss_out);

    {   // graph user CL
        int nb16 = U / 16;
        infonce_wmma_kernel<<<(nb16 + 7) / 8, 256, 0, stream>>>(
            graph_user_emb, graph_user_emb_cl, neg_user, nb16, 1.0f / (float)U, loss_out);
    }
    {   // graph item CL
        int nb16 = I / 16;
        infonce_wmma_kernel<<<(nb16 + 7) / 8, 256, 0, stream>>>(
            graph_item_emb, graph_item_emb_cl, neg_item, nb16, 1.0f / (float)I, loss_out);
    }
    {   // sequence CL
        int nb16 = B / 16;
        infonce_wmma_kernel<<<(nb16 + 7) / 8, 256, 0, stream>>>(
            seq_last_emb, seq_last_emb_cl, neg_seq, nb16, 1.0f / (float)B, loss_out);
    }
    {   // user fusion
        int nvec = B * (EMB_D / 4);
        fusion_user_kernel<<<(nvec + 255) / 256, 256, 0, stream>>>(
            graph_user_emb, seq_last_emb, user_attn_w, user, user_out, nvec);
    }
    {   // item fusion
        int nvec = I * (EMB_D / 4);
        fusion_item_kernel<<<(nvec + 255) / 256, 256, 0, stream>>>(
            graph_item_emb, seq_item_emb, item_attn_w, item_out, nvec);
    }
}